// CustomWeightedTensorProduct_62294205661793
// MI455X (gfx1250) — compile-verified
//
#include <hip/hip_runtime.h>
#include <stdint.h>

#define ZPB 4   // z-rows per block; 256 threads = 4 rows x 64 channels = 8 wave32

// ---- constants (real-basis Wigner 3j, scaled by sqrt(2*lo+1)) ----
#define S2    0.70710678118654752f   // 1/sqrt2
#define S3    0.57735026918962576f   // 1/sqrt3
#define S5    0.44721359549995794f   // 1/sqrt5
#define S6    0.40824829046386302f   // 1/sqrt6
#define S6H   0.20412414523193151f   // 1/(2 sqrt6)
#define S2H   0.35355339059327376f   // 1/(2 sqrt2)
#define S10   0.31622776601683794f   // 1/sqrt10
#define SQ35  0.77459666924148338f   // sqrt(3/5)
#define SR3   1.73205080756887729f   // sqrt3
#define C222F 1.30930734141595430f   // 2*sqrt(3/7)
#define INV_SQRT3 0.57735026918962576f
#define INV_SQRT6 0.40824829046386302f

// Streamed-once data: non-temporal load/store so the 278 MB single-pass
// stream does not evict reusable lines from the 192 MB L2.
#define NTL(p)     __builtin_nontemporal_load(p)
#define NTS(v, p)  __builtin_nontemporal_store((v), (p))

__global__ __launch_bounds__(256)
void ctp_kernel(const float* __restrict__ x1,
                const float* __restrict__ x2,
                const float* __restrict__ w,
                float* __restrict__ out, int Z)
{
    __shared__ float shx2[ZPB * 9];
    const int tid   = threadIdx.x;
    const int zbase = blockIdx.x * ZPB;

    // ---- CDNA5 async global->LDS staging of the broadcast x2 rows ----
    if (tid < ZPB * 9) {
        const int idx = zbase * 9 + tid;
        if (idx < Z * 9) {
            uint64_t ga = (uint64_t)(uintptr_t)(x2 + idx);
            // flat LDS addresses carry the LDS byte offset in the low 32 bits
            uint32_t la = (uint32_t)(uintptr_t)(&shx2[tid]);
            asm volatile("global_load_async_to_lds_b32 %0, %1, off"
                         :: "v"(la), "v"(ga) : "memory");
        }
    }
    asm volatile("s_wait_asynccnt 0x0" ::: "memory");
    __syncthreads();

    const int zi = tid >> 6;
    const int uu = tid & 63;
    const int z  = zbase + zi;
    if (z >= Z) return;

    const float* __restrict__ x1r  = x1 + (size_t)z * 576;
    const float* __restrict__ wr   = w  + (size_t)z * 960;
    float*       __restrict__ outr = out + (size_t)z * 576;
    const float* b = &shx2[zi * 9];   // wave-uniform LDS broadcast reads

    // ---- inputs (read-once -> NT) ----
    const float a0 = NTL(x1r + uu);
    const float u0 = NTL(x1r + 64 + 3*uu + 0);
    const float u1 = NTL(x1r + 64 + 3*uu + 1);
    const float u2 = NTL(x1r + 64 + 3*uu + 2);
    const float q0 = NTL(x1r + 256 + 5*uu + 0);
    const float q1 = NTL(x1r + 256 + 5*uu + 1);
    const float q2 = NTL(x1r + 256 + 5*uu + 2);
    const float q3 = NTL(x1r + 256 + 5*uu + 3);
    const float q4 = NTL(x1r + 256 + 5*uu + 4);

    const float b0 = b[0];
    const float v0 = b[1], v1 = b[2], v2 = b[3];
    const float r0 = b[4], r1 = b[5], r2 = b[6], r3 = b[7], r4 = b[8];

    const float w0  = NTL(wr +  0*64 + uu), w1  = NTL(wr +  1*64 + uu);
    const float w2  = NTL(wr +  2*64 + uu), w3  = NTL(wr +  3*64 + uu);
    const float w4  = NTL(wr +  4*64 + uu), w5  = NTL(wr +  5*64 + uu);
    const float w6  = NTL(wr +  6*64 + uu), w7  = NTL(wr +  7*64 + uu);
    const float w8  = NTL(wr +  8*64 + uu), w9  = NTL(wr +  9*64 + uu);
    const float w10 = NTL(wr + 10*64 + uu), w11 = NTL(wr + 11*64 + uu);
    const float w12 = NTL(wr + 12*64 + uu), w13 = NTL(wr + 13*64 + uu);
    const float w14 = NTL(wr + 14*64 + uu);

    // ---- accumulators ----
    float o0 = 0.f;
    float e0 = 0.f, e1 = 0.f, e2 = 0.f;                       // lo = 1
    float f0 = 0.f, f1 = 0.f, f2 = 0.f, f3 = 0.f, f4 = 0.f;   // lo = 2

    // p0 (0,0,0)
    o0 += w0 * (a0 * b0);
    // p4 (1,1,0): delta/sqrt3
    o0 += w4 * S3 * (u0*v0 + u1*v1 + u2*v2);
    // p12 (2,2,0): delta/sqrt5
    o0 += w12 * S5 * (q0*r0 + q1*r1 + q2*r2 + q3*r3 + q4*r4);

    // p1 (0,1,1): identity
    { float s = w1 * a0; e0 += s*v0; e1 += s*v1; e2 += s*v2; }
    // p3 (1,0,1): identity
    { float s = w3 * b0; e0 += s*u0; e1 += s*u1; e2 += s*u2; }
    // p5 (1,1,1): eps/sqrt2 (cyclic (y,z,x) basis)
    { float s = w5 * S2;
      e0 += s*(u1*v2 - u2*v1);
      e1 += s*(u2*v0 - u0*v2);
      e2 += s*(u0*v1 - u1*v0); }
    // p7 (1,2,1): sqrt(3/5) * M(r) u, M = sum_J r_J A_J (sym traceless basis)
    { const float M00 = -S6*r2 - S2*r4, M01 = S2*r1, M02 = S2*r0;
      const float M11 = 2.f*S6*r2,      M12 = S2*r3, M22 = -S6*r2 + S2*r4;
      float s = w7 * SQ35;
      e0 += s*(M00*u0 + M01*u1 + M02*u2);
      e1 += s*(M01*u0 + M11*u1 + M12*u2);
      e2 += s*(M02*u0 + M12*u1 + M22*u2); }
    // p10 (2,1,1): sqrt(3/5) * M(q) v
    { const float N00 = -S6*q2 - S2*q4, N01 = S2*q1, N02 = S2*q0;
      const float N11 = 2.f*S6*q2,      N12 = S2*q3, N22 = -S6*q2 + S2*q4;
      float s = w10 * SQ35;
      e0 += s*(N00*v0 + N01*v1 + N02*v2);
      e1 += s*(N01*v0 + N11*v1 + N12*v2);
      e2 += s*(N02*v0 + N12*v1 + N22*v2); }
    // p13 (2,2,1): antisymmetric coupling / sqrt10
    { float s = w13 * S10;
      e0 += s*( q0*r1 - q1*r0 - SR3*(q2*r3 - q3*r2) - q3*r4 + q4*r3 );
      e1 += s*( 2.f*(q0*r4 - q4*r0) + q1*r3 - q3*r1 );
      e2 += s*( q3*r0 - q0*r3 - SR3*(q1*r2 - q2*r1) + q1*r4 - q4*r1 ); }

    // p2 (0,2,2): identity
    { float s = w2 * a0; f0 += s*r0; f1 += s*r1; f2 += s*r2; f3 += s*r3; f4 += s*r4; }
    // p9 (2,0,2): identity
    { float s = w9 * b0; f0 += s*q0; f1 += s*q1; f2 += s*q2; f3 += s*q3; f4 += s*q4; }
    // p6 (1,1,2): symmetric traceless outer product
    { f0 += w6 * S2 * (u0*v2 + u2*v0);
      f1 += w6 * S2 * (u0*v1 + u1*v0);
      f2 += w6 * S6 * (2.f*u1*v1 - u0*v0 - u2*v2);
      f3 += w6 * S2 * (u1*v2 + u2*v1);
      f4 += w6 * S2 * (u2*v2 - u0*v0); }
    // p8 (1,2,2): antisymmetric 1x2->2 coupling
    { f0 += w8 * S6 * ( u2*r3 - u0*r1 - 2.f*u1*r4 );
      f1 += w8 * ( S6*(u0*r0 - u1*r3 + u2*r4) + S2*(u2*r2) );
      f2 += w8 * S2 * ( u0*r3 - u2*r1 );
      f3 += w8 * ( S6*(u0*r4 + u1*r1 - u2*r0) - S2*(u0*r2) );
      f4 += w8 * S6 * ( 2.f*u1*r0 - u0*r3 - u2*r1 ); }
    // p11 (2,1,2): same invariant, slots swapped (u->v, r->q)
    { f0 += w11 * S6 * ( v2*q3 - v0*q1 - 2.f*v1*q4 );
      f1 += w11 * ( S6*(v0*q0 - v1*q3 + v2*q4) + S2*(v2*q2) );
      f2 += w11 * S2 * ( v0*q3 - v2*q1 );
      f3 += w11 * ( S6*(v0*q4 + v1*q1 - v2*q0) - S2*(v0*q2) );
      f4 += w11 * S6 * ( 2.f*v1*q0 - v0*q3 - v2*q1 ); }
    // p14 (2,2,2): fully symmetric coupling, -2*sqrt(3/7)*g
    { float s = -w14 * C222F;
      f0 += s*( S2H*(q1*r3 + q3*r1) - S6*(q0*r2 + q2*r0) );
      f1 += s*( S6H*(q1*r2 + q2*r1) + S2H*(q0*r3 + q3*r0 - q1*r4 - q4*r1) );
      f2 += s*S6*( q2*r2 - q4*r4 - q0*r0 + 0.5f*(q1*r1 + q3*r3) );
      f3 += s*( S6H*(q3*r2 + q2*r3) + S2H*(q3*r4 + q4*r3 + q0*r1 + q1*r0) );
      f4 += s*( S2H*(q3*r3 - q1*r1) - S6*(q2*r4 + q4*r2) ); }

    // ---- normalize (1/sqrt(cnt)) and store (write-once -> NT) ----
    NTS(o0 * INV_SQRT3, outr + uu);
    NTS(e0 * INV_SQRT6, outr + 64 + 3*uu + 0);
    NTS(e1 * INV_SQRT6, outr + 64 + 3*uu + 1);
    NTS(e2 * INV_SQRT6, outr + 64 + 3*uu + 2);
    NTS(f0 * INV_SQRT6, outr + 256 + 5*uu + 0);
    NTS(f1 * INV_SQRT6, outr + 256 + 5*uu + 1);
    NTS(f2 * INV_SQRT6, outr + 256 + 5*uu + 2);
    NTS(f3 * INV_SQRT6, outr + 256 + 5*uu + 3);
    NTS(f4 * INV_SQRT6, outr + 256 + 5*uu + 4);
}

extern "C" void kernel_launch(void* const* d_in, const int* in_sizes, int n_in,
                              void* d_out, int out_size, void* d_ws, size_t ws_size,
                              hipStream_t stream) {
    (void)n_in; (void)out_size; (void)d_ws; (void)ws_size;
    const float* x1 = (const float*)d_in[0];
    const float* x2 = (const float*)d_in[1];
    const float* w  = (const float*)d_in[2];
    float* out = (float*)d_out;
    const int Z = in_sizes[1] / 9;
    const int blocks = (Z + ZPB - 1) / ZPB;
    hipLaunchKernelGGL(ctp_kernel, dim3(blocks), dim3(256), 0, stream,
                       x1, x2, w, out, Z);
}